// skipfeature_fusion_88046829568384
// MI455X (gfx1250) — compile-verified
//
#include <hip/hip_runtime.h>
#include <hip/hip_bf16.h>
#include <stdint.h>

// B=8, C=64, H=256, W=256 fp32 -> out [B,128,H,W]. Pure streaming op:
// 512 MiB total traffic => ~23 us floor @ 23.3 TB/s HBM; instruction issue is
// NOT the limiter (round-1 mix ~4.4 us of issue), so we keep the round-1
// shape (TILE_W=128 => 512B LDS rows => ds_load_2addr pairing + VOPD) and add
// TDM double-buffering so the DMA engine always has a tile pair in flight.

#define HW_       65536   // H*W elements per channel plane
#define TILE_W    128     // pixels per tile (half a row)
#define NCH       64
#define NTHREADS  128     // 4 wave32, 1 px/thread
#define TILES_PB  4       // tiles per block (double-buffered pipeline)

typedef unsigned int v4u __attribute__((ext_vector_type(4)));
typedef int          v4i __attribute__((ext_vector_type(4)));
typedef int          v8i __attribute__((ext_vector_type(8)));

// TDM 2D tile: tile_dim0 = TILE_W contiguous floats, tile_dim1 = NCH lines
// strided by HW_ elements. One tensor_load_to_lds per call.
__device__ __forceinline__ void tdm_load_tile(const float* gptr, unsigned lds_off) {
    unsigned long long ga = (unsigned long long)(uintptr_t)gptr;

    v4u g0;
    g0.x = 1u;                                                // count=1 (valid)
    g0.y = lds_off;                                           // LDS byte address
    g0.z = (unsigned)(ga & 0xFFFFFFFFu);                      // global_addr[31:0]
    g0.w = ((unsigned)(ga >> 32) & 0x01FFFFFFu) | (2u << 30); // addr[56:32] | type=2

    v8i g1;
    g1[0] = 0x00020000;                      // data_size=2 (4B), everything else off
    g1[1] = (int)((unsigned)TILE_W << 16);   // tensor_dim0 lo16
    g1[2] = (int)((unsigned)NCH    << 16);   // tdim0 hi16=0 | tensor_dim1 lo16
    g1[3] = (int)((unsigned)TILE_W << 16);   // tdim1 hi16=0 | tile_dim0
    g1[4] = NCH;                             // tile_dim1 | tile_dim2=0
    g1[5] = HW_;                             // tensor_dim0_stride lo32 (elements)
    g1[6] = 0;
    g1[7] = 0;

    v4i z4 = {0, 0, 0, 0};
#if __clang_major__ >= 23
    v8i z8 = {0, 0, 0, 0, 0, 0, 0, 0};
    __builtin_amdgcn_tensor_load_to_lds(g0, g1, z4, z4, z8, 0);
#else
    __builtin_amdgcn_tensor_load_to_lds(g0, g1, z4, z4, 0);
#endif
}

__device__ __forceinline__ float sigmoidf_(float t) {
    return 1.0f / (1.0f + __expf(-t));
}

__global__ __launch_bounds__(NTHREADS) void skipfeature_fusion_kernel(
    const float* __restrict__ x1, const float* __restrict__ x2,
    float* __restrict__ out) {
    // [buffer][tensor][channel][pixel] : 2*2*64*128*4 = 128 KB
    __shared__ __align__(16) float tile[2][2][NCH][TILE_W];

    const unsigned t0 = blockIdx.x * TILES_PB;

    // Prologue: DMA tile 0 pair into buffer 0 (wave 0 drives all DMAs).
    if (threadIdx.x < 32) {
        const unsigned t = t0;
        const unsigned w0 = (t & 1u) * TILE_W, h = (t >> 1) & 255u, b = t >> 9;
        const size_t in_base = (size_t)b * NCH * HW_ + (size_t)h * 256 + w0;
        tdm_load_tile(x1 + in_base, (unsigned)(uintptr_t)&tile[0][0][0][0]);
        tdm_load_tile(x2 + in_base, (unsigned)(uintptr_t)&tile[0][1][0][0]);
    }

    const int px = (int)threadIdx.x;

    for (int i = 0; i < TILES_PB; ++i) {
        const int buf = i & 1;

        __syncthreads();   // readers of buffer (i+1)&1 (iter i-1) have drained
        if (threadIdx.x < 32) {
            if (i + 1 < TILES_PB) {     // prefetch next tile pair into spare buffer
                const unsigned t = t0 + i + 1;
                const unsigned w0 = (t & 1u) * TILE_W, h = (t >> 1) & 255u, b = t >> 9;
                const size_t in_base = (size_t)b * NCH * HW_ + (size_t)h * 256 + w0;
                const int nb = (i + 1) & 1;
                tdm_load_tile(x1 + in_base, (unsigned)(uintptr_t)&tile[nb][0][0][0]);
                tdm_load_tile(x2 + in_base, (unsigned)(uintptr_t)&tile[nb][1][0][0]);
                // tensor ops of one wave complete in order: <=2 outstanding
                // means tile i's pair has landed.
                __builtin_amdgcn_s_wait_tensorcnt(2);
            } else {
                __builtin_amdgcn_s_wait_tensorcnt(0);
            }
        }
        __syncthreads();   // publish tile i's LDS to all waves

        const float (*src1)[TILE_W] = tile[buf][0];
        const float (*src2)[TILE_W] = tile[buf][1];

        // ---- Pass 1: group sum/max over channels. Group g <- channels c with
        // (c-1)&3 == g, c = 1..63 (group 3 has only 15 members: 4..60).
        float s1[4] = {0.f, 0.f, 0.f, 0.f};
        float s2[4] = {0.f, 0.f, 0.f, 0.f};
        float m1[4] = {-INFINITY, -INFINITY, -INFINITY, -INFINITY};
        float m2[4] = {-INFINITY, -INFINITY, -INFINITY, -INFINITY};

#pragma unroll
        for (int c = 1; c < NCH; ++c) {
            const int g = (c - 1) & 3;
            const float v1 = src1[c][px];
            const float v2 = src2[c][px];
            s1[g] += v1; m1[g] = fmaxf(m1[g], v1);
            s2[g] += v2; m2[g] = fmaxf(m2[g], v2);
        }

        const float inv_cnt[4] = {1.f / 16.f, 1.f / 16.f, 1.f / 16.f, 1.f / 15.f};
        float w1[4], w2[4];
#pragma unroll
        for (int g = 0; g < 4; ++g) {
            w1[g] = sigmoidf_(s1[g] * inv_cnt[g] + m1[g]);
            w2[g] = sigmoidf_(s2[g] * inv_cnt[g] + m2[g]);
        }

        // ---- Pass 2: out[b,c] = w1[c&3]*x2[b,c]; out[b,64+c] = w2[c&3]*x1[b,c]
        const unsigned t = t0 + i;
        const unsigned w0 = (t & 1u) * TILE_W, h = (t >> 1) & 255u, b = t >> 9;
        const size_t out_base =
            (size_t)b * (2 * NCH) * HW_ + (size_t)h * 256 + w0 + px;
#pragma unroll
        for (int c = 0; c < NCH; ++c) {
            const int gi = c & 3;
            const float r0 = w1[gi] * src2[c][px];
            const float r1 = w2[gi] * src1[c][px];
            __builtin_nontemporal_store(r0, out + out_base + (size_t)c * HW_);
            __builtin_nontemporal_store(r1, out + out_base + (size_t)(NCH + c) * HW_);
        }
    }
}

extern "C" void kernel_launch(void* const* d_in, const int* in_sizes, int n_in,
                              void* d_out, int out_size, void* d_ws, size_t ws_size,
                              hipStream_t stream) {
    (void)in_sizes; (void)n_in; (void)d_ws; (void)ws_size; (void)out_size;
    const float* x1 = (const float*)d_in[0];
    const float* x2 = (const float*)d_in[1];
    float* out = (float*)d_out;

    const int B = 8, H = 256, W = 256;
    const int n_tiles = B * H * (W / TILE_W);        // 4096
    dim3 grid(n_tiles / TILES_PB);                   // 1024 blocks
    dim3 block(NTHREADS);                            // 128 threads = 4 wave32
    skipfeature_fusion_kernel<<<grid, block, 0, stream>>>(x1, x2, out);
}